// GCNStepPredictor_19121194402500
// MI455X (gfx1250) — compile-verified
//
#include <hip/hip_runtime.h>
#include <hip/hip_bf16.h>

typedef __attribute__((ext_vector_type(2))) float v2f;
typedef __attribute__((ext_vector_type(8))) float v8f;

#define HDIM 128
#define TILE_ROWS 64
#define A_STRIDE 132   // pad LDS rows to dodge bank conflicts
#define KSTEPS (HDIM / 4)

// ---------------------------------------------------------------------------
// Degree: deg[dst] += 1 per edge
// ---------------------------------------------------------------------------
__global__ __launch_bounds__(256) void deg_kernel(const int* __restrict__ dst,
                                                  float* __restrict__ deg, int E) {
    int e = blockIdx.x * 256 + threadIdx.x;
    if (e < E) unsafeAtomicAdd(&deg[dst[e]], 1.0f);
}

// deg (with self loop) -> inv_sqrt_deg, inv_deg (in place over deg)
__global__ __launch_bounds__(256) void norm_kernel(float* __restrict__ deg_io,
                                                   float* __restrict__ inv_sqrt,
                                                   int n) {
    int i = blockIdx.x * 256 + threadIdx.x;
    if (i < n) {
        float d = deg_io[i] + 1.0f;
        inv_sqrt[i] = rsqrtf(d);
        deg_io[i] = 1.0f / d;   // becomes inv_deg
    }
}

// ---------------------------------------------------------------------------
// GEMM: C[n,128] = A[n,128] @ W[128,128] via V_WMMA_F32_16X16X4_F32
// Block: 256 thr (8 waves). Each wave owns cols [16w,16w+16); block does 64 rows.
// B fragments are hoisted into registers once per wave and reused across the
// 4 row-tiles -> inner loop is 1 ds_load_b64 + 1 v_wmma per K-step.
// ---------------------------------------------------------------------------
__global__ __launch_bounds__(256) void gemm_wmma_kernel(const float* __restrict__ A,
                                                        const float* __restrict__ W,
                                                        float* __restrict__ C,
                                                        int nrows) {
    __shared__ float Wsh[HDIM * HDIM];           // 64 KB
    __shared__ float Ash[TILE_ROWS * A_STRIDE];  // ~33 KB (padded rows)

    const int tid  = threadIdx.x;
    const int wave = tid >> 5;
    const int lane = tid & 31;
    const int lane16   = lane & 15;
    const int halfsel  = lane >> 4;          // 0: lanes 0-15, 1: lanes 16-31
    const int rowbase  = blockIdx.x * TILE_ROWS;

    // Stage W (row-major 128x128), coalesced float4
    for (int i = tid; i < HDIM * HDIM / 4; i += 256)
        ((float4*)Wsh)[i] = ((const float4*)W)[i];

    // Stage A tile (64x128), zero-fill rows past nrows; padded LDS stride
    for (int i = tid; i < TILE_ROWS * HDIM / 4; i += 256) {
        int r  = i >> 5;        // / (128/4)
        int c4 = i & 31;
        float4 v = make_float4(0.f, 0.f, 0.f, 0.f);
        int gr = rowbase + r;
        if (gr < nrows) v = ((const float4*)A)[(size_t)gr * (HDIM / 4) + c4];
        *(float4*)&Ash[r * A_STRIDE + c4 * 4] = v;
    }
    __syncthreads();

    const int colbase = wave * 16;

    // Hoist B fragments: B frag layout -> lane16 = N col;
    // VGPR j holds K=j (lanes 0-15) / K=j+2 (lanes 16-31)
    v2f bfrag[KSTEPS];
#pragma unroll
    for (int ks = 0; ks < KSTEPS; ++ks) {
        int kk = ks * 4 + halfsel * 2;
        bfrag[ks][0] = Wsh[(kk + 0) * HDIM + colbase + lane16];
        bfrag[ks][1] = Wsh[(kk + 1) * HDIM + colbase + lane16];
    }

    for (int rt = 0; rt < TILE_ROWS / 16; ++rt) {
        v8f acc = {0.f, 0.f, 0.f, 0.f, 0.f, 0.f, 0.f, 0.f};
        const float* arow = &Ash[(rt * 16 + lane16) * A_STRIDE];
        // A frag: lane16 = M; VGPR0/1 = K, K+1; lanes 16-31 shifted by +2 in K
#pragma unroll
        for (int ks = 0; ks < KSTEPS; ++ks) {
            int kk = ks * 4 + halfsel * 2;
            v2f a = *(const v2f*)&arow[kk];   // 8B LDS load (ds_load_b64)
            acc = __builtin_amdgcn_wmma_f32_16x16x4_f32(
                      false, a, false, bfrag[ks], (short)0, acc, false, false);
        }
        // C/D layout: VGPR j -> M = j (lanes 0-15) or j+8 (lanes 16-31); N = lane16
        int gcol = colbase + lane16;
        int grow0 = rowbase + rt * 16 + halfsel * 8;
#pragma unroll
        for (int j = 0; j < 8; ++j) {
            int gr = grow0 + j;
            if (gr < nrows) C[(size_t)gr * HDIM + gcol] = acc[j];
        }
    }
}

// ---------------------------------------------------------------------------
// Edge scatter: agg[dst] += hw[src] * inv_sqrt[src]*inv_sqrt[dst]
// 32 lanes per edge, float4 per lane, fp32 hardware atomics
// ---------------------------------------------------------------------------
__global__ __launch_bounds__(256) void scatter_kernel(const float* __restrict__ hw,
                                                      const int* __restrict__ src,
                                                      const int* __restrict__ dst,
                                                      const float* __restrict__ inv_sqrt,
                                                      float* __restrict__ agg, int E) {
    int tid = blockIdx.x * 256 + threadIdx.x;
    int e = tid >> 5;
    if (e >= E) return;
    int f4 = tid & 31;                     // lane handles floats [4*f4, 4*f4+4)
    int s = src[e], d = dst[e];
    float norm = inv_sqrt[s] * inv_sqrt[d];
    float4 m = ((const float4*)(hw + (size_t)s * HDIM))[f4];
    float* p = agg + (size_t)d * HDIM + f4 * 4;
    unsafeAtomicAdd(p + 0, m.x * norm);
    unsafeAtomicAdd(p + 1, m.y * norm);
    unsafeAtomicAdd(p + 2, m.z * norm);
    unsafeAtomicAdd(p + 3, m.w * norm);
}

// ---------------------------------------------------------------------------
// Fused epilogue per node: agg + hw*inv_deg + bias -> LayerNorm -> (GELU) ->
// hout; on last layer also scores = h . head_w + head_b
// ---------------------------------------------------------------------------
__global__ __launch_bounds__(128) void post_kernel(const float* __restrict__ agg,
                                                   const float* __restrict__ hw,
                                                   const float* __restrict__ inv_deg,
                                                   const float* __restrict__ bias,
                                                   const float* __restrict__ ln_g,
                                                   const float* __restrict__ ln_b,
                                                   const float* __restrict__ head_w,
                                                   const float* __restrict__ head_b,
                                                   float* __restrict__ hout,
                                                   float* __restrict__ scores,
                                                   int apply_gelu) {
    const int node = blockIdx.x;
    const int f = threadIdx.x;
    __shared__ float red[HDIM];
    size_t idx = (size_t)node * HDIM + f;

    float v = agg[idx] + hw[idx] * inv_deg[node] + bias[f];

    red[f] = v; __syncthreads();
#pragma unroll
    for (int s = 64; s > 0; s >>= 1) { if (f < s) red[f] += red[f + s]; __syncthreads(); }
    float mu = red[0] * (1.0f / HDIM);
    __syncthreads();

    float dlt = v - mu;
    red[f] = dlt * dlt; __syncthreads();
#pragma unroll
    for (int s = 64; s > 0; s >>= 1) { if (f < s) red[f] += red[f + s]; __syncthreads(); }
    float var = red[0] * (1.0f / HDIM);
    __syncthreads();

    float hv = dlt * rsqrtf(var + 1e-5f) * ln_g[f] + ln_b[f];
    if (apply_gelu)
        hv = 0.5f * hv * (1.0f + erff(hv * 0.70710678118654752f));
    hout[idx] = hv;

    if (scores != nullptr) {
        red[f] = hv * head_w[f]; __syncthreads();
#pragma unroll
        for (int s = 64; s > 0; s >>= 1) { if (f < s) red[f] += red[f + s]; __syncthreads(); }
        if (f == 0) scores[node] = red[0] + head_b[0];
    }
}

// ---------------------------------------------------------------------------
extern "C" void kernel_launch(void* const* d_in, const int* in_sizes, int n_in,
                              void* d_out, int out_size, void* d_ws, size_t ws_size,
                              hipStream_t stream) {
    const float* x       = (const float*)d_in[0];
    const int*   edges   = (const int*)d_in[1];
    const float* conv_w  = (const float*)d_in[2];
    const float* conv_b  = (const float*)d_in[3];
    const float* ln_g    = (const float*)d_in[4];
    const float* ln_b    = (const float*)d_in[5];
    const float* head_w  = (const float*)d_in[6];
    const float* head_b  = (const float*)d_in[7];

    const int N = in_sizes[0] / HDIM;
    const int E = in_sizes[1] / 2;
    const int L = in_sizes[3] / HDIM;   // conv_b is (L, H)

    const int* src = edges;
    const int* dst = edges + E;

    float* scores = (float*)d_out;        // first N floats
    float* h_fin  = (float*)d_out + N;    // then N*H floats

    // workspace layout (floats)
    float* wsf       = (float*)d_ws;
    float* inv_deg   = wsf;                       // N (deg -> inv_deg in place)
    float* inv_sqrt  = wsf + N;                   // N
    float* hwbuf     = wsf + 2 * (size_t)N;                       // N*H
    float* aggbuf    = hwbuf + (size_t)N * HDIM;                  // N*H
    float* hA        = aggbuf + (size_t)N * HDIM;                 // N*H
    float* hB        = hA + (size_t)N * HDIM;                     // N*H

    // degrees + normalization factors
    hipMemsetAsync(inv_deg, 0, (size_t)N * sizeof(float), stream);
    deg_kernel<<<(E + 255) / 256, 256, 0, stream>>>(dst, inv_deg, E);
    norm_kernel<<<(N + 255) / 256, 256, 0, stream>>>(inv_deg, inv_sqrt, N);

    const int gemm_blocks = (N + TILE_ROWS - 1) / TILE_ROWS;
    const int scat_blocks = (int)(((long long)E * 32 + 255) / 256);

    const float* hin = x;
    for (int i = 0; i < L; ++i) {
        float* hout = (i == L - 1) ? h_fin : ((i & 1) ? hB : hA);

        gemm_wmma_kernel<<<gemm_blocks, 256, 0, stream>>>(
            hin, conv_w + (size_t)i * HDIM * HDIM, hwbuf, N);

        hipMemsetAsync(aggbuf, 0, (size_t)N * HDIM * sizeof(float), stream);
        scatter_kernel<<<scat_blocks, 256, 0, stream>>>(
            hwbuf, src, dst, inv_sqrt, aggbuf, E);

        post_kernel<<<N, HDIM, 0, stream>>>(
            aggbuf, hwbuf, inv_deg,
            conv_b + (size_t)i * HDIM, ln_g + (size_t)i * HDIM, ln_b + (size_t)i * HDIM,
            head_w, head_b,
            hout, (i == L - 1) ? scores : nullptr,
            (i != L - 1) ? 1 : 0);

        hin = hout;
    }
}